// SSM_10737418240374
// MI455X (gfx1250) — compile-verified
//
#include <hip/hip_runtime.h>
#include <math.h>

#define BATCH   8
#define DMODEL  192
#define HROWS   32
#define WCOLS   32
#define LSEQ    1024
#define DINNER  384
#define DHALF   192
#define DSTATE  128
#define DTRANK  12
#define RTOT    268   /* DT_RANK + 2*D_STATE */
#define RPAD    272   /* padded to 16 */

typedef float v2f __attribute__((ext_vector_type(2)));
typedef float v8f __attribute__((ext_vector_type(8)));

__device__ __forceinline__ v8f wmma_f32_step(v2f a, v2f b, v8f c) {
  // D = A(16x4,f32) * B(4x16,f32) + C(16x16,f32)
  return __builtin_amdgcn_wmma_f32_16x16x4_f32(false, a, false, b, (short)0, c,
                                               false, false);
}

// ---------------------------------------------------------------------------
// K1: serpentine gather + xz = hs_serp @ W_in^T, output layout (B, DINNER, L)
// One wave per 16x16 tile; K = 192 -> 48 chained v_wmma_f32_16x16x4_f32.
// ---------------------------------------------------------------------------
__global__ void __launch_bounds__(128) k_gemm_in(const float* __restrict__ hs,
                                                 const float* __restrict__ Win,
                                                 float* __restrict__ xz) {
  const int lane = threadIdx.x & 31;
  const int wv   = threadIdx.x >> 5;
  const int tm   = blockIdx.x * 4 + wv;   // 0..23   (M = 384)
  const int tn   = blockIdx.y;            // 0..63   (N = 1024)
  const int b    = blockIdx.z;
  const int m    = lane & 15;
  const int kh   = lane >> 4;

  const int arow = tm * 16 + m;           // output channel (row of W_in)
  const int l    = tn * 16 + m;           // serpentine-space column
  const int hr   = l >> 5, wc = l & 31;
  const int wsrc = (hr & 1) ? (31 - wc) : wc;   // undo row flip on load
  const float* bbase = hs + ((size_t)b * DMODEL) * LSEQ + hr * WCOLS + wsrc;

  v8f acc = {0.f, 0.f, 0.f, 0.f, 0.f, 0.f, 0.f, 0.f};
  for (int k = 0; k < DMODEL; k += 4) {
    const int kk = k + 2 * kh;
    v2f a = *(const v2f*)(Win + (size_t)arow * DMODEL + kk);   // 8B aligned
    v2f bf;
    bf.x = bbase[(size_t)kk * LSEQ];
    bf.y = bbase[(size_t)(kk + 1) * LSEQ];
    acc = wmma_f32_step(a, bf, acc);
  }
  float* dst = xz + ((size_t)b * DINNER) * LSEQ + tn * 16 + m;
  const int mofs = 8 * kh;
#pragma unroll
  for (int v = 0; v < 8; ++v)
    dst[(size_t)(tm * 16 + v + mofs) * LSEQ] = acc[v];
}

// ---------------------------------------------------------------------------
// K2: causal depthwise conv (k=4) + SiLU for x (-> xconv) and z (-> ycat hi)
// ---------------------------------------------------------------------------
__global__ void __launch_bounds__(256) k_conv_silu(const float* __restrict__ xz,
                                                   const float* __restrict__ wx,
                                                   const float* __restrict__ wz,
                                                   float* __restrict__ xconv,
                                                   float* __restrict__ ycat) {
  const size_t idx = (size_t)blockIdx.x * blockDim.x + threadIdx.x;
  const int l = (int)(idx & (LSEQ - 1));
  const int d = (int)((idx >> 10) % DINNER);
  const int b = (int)(idx / ((size_t)DINNER * LSEQ));
  const float* src = xz + (idx - (size_t)l);          // channel-row base
  const float* w = (d < DHALF) ? (wx + d * 4) : (wz + (d - DHALF) * 4);
  float s = 0.f;
#pragma unroll
  for (int i = 0; i < 4; ++i) {
    const int li = l + i - 3;
    const float xv = (li >= 0) ? src[li] : 0.f;
    s = fmaf(w[i], xv, s);
  }
  const float y = s / (1.f + __expf(-s));             // SiLU
  if (d < DHALF) xconv[((size_t)b * DHALF + d) * LSEQ + l] = y;
  else           ycat[((size_t)b * DINNER + d) * LSEQ + l] = y;
}

// ---------------------------------------------------------------------------
// K3: x_dbl[b,l,r] = sum_d W_xproj[r,d] * xconv[b,d,l]; layout (B, L, RPAD)
// M = 268 (padded to 272 -> 17 tiles), K = 192.
// ---------------------------------------------------------------------------
__global__ void __launch_bounds__(32) k_gemm_xproj(const float* __restrict__ xconv,
                                                   const float* __restrict__ Wxp,
                                                   float* __restrict__ xdbl) {
  const int lane = threadIdx.x & 31;
  const int tm = blockIdx.x;              // 0..16
  const int tn = blockIdx.y;              // 0..63
  const int b  = blockIdx.z;
  const int m = lane & 15, kh = lane >> 4;
  const int arow = tm * 16 + m;           // 0..271
  const int l = tn * 16 + m;
  const float* bbase = xconv + ((size_t)b * DHALF) * LSEQ + l;

  v8f acc = {0.f, 0.f, 0.f, 0.f, 0.f, 0.f, 0.f, 0.f};
  for (int k = 0; k < DHALF; k += 4) {
    const int kk = k + 2 * kh;
    v2f a;
    if (arow < RTOT) a = *(const v2f*)(Wxp + (size_t)arow * DHALF + kk);
    else             { a.x = 0.f; a.y = 0.f; }
    v2f bf;
    bf.x = bbase[(size_t)kk * LSEQ];
    bf.y = bbase[(size_t)(kk + 1) * LSEQ];
    acc = wmma_f32_step(a, bf, acc);
  }
  float* dst = xdbl + ((size_t)b * LSEQ + l) * RPAD;
#pragma unroll
  for (int v = 0; v < 8; ++v) {
    const int r = tm * 16 + v + 8 * kh;
    if (r < RTOT) dst[r] = acc[v];
  }
}

// ---------------------------------------------------------------------------
// K4: delta = softplus(dt @ W_dt^T + 2*b_dt)   (reference adds b_dt twice)
// K = 12 -> scalar FMAs, one thread per (b,d,l).
// ---------------------------------------------------------------------------
__global__ void __launch_bounds__(256) k_dt_delta(const float* __restrict__ xdbl,
                                                  const float* __restrict__ Wdt,
                                                  const float* __restrict__ bdt,
                                                  float* __restrict__ delta) {
  const size_t idx = (size_t)blockIdx.x * blockDim.x + threadIdx.x;
  const int l = (int)(idx & (LSEQ - 1));
  const int d = (int)((idx >> 10) % DHALF);
  const int b = (int)(idx / ((size_t)DHALF * LSEQ));
  const float* xr = xdbl + ((size_t)b * LSEQ + l) * RPAD;
  float s = 2.f * bdt[d];
#pragma unroll
  for (int r = 0; r < DTRANK; ++r) s = fmaf(xr[r], Wdt[d * DTRANK + r], s);
  delta[idx] = fmaxf(s, 0.f) + log1pf(__expf(-fabsf(s)));   // stable softplus
}

// ---------------------------------------------------------------------------
// K5: selective scan. One wave per (d,b); 4 states per lane (n = lane + 32j).
// y[b,d,t] = sum_n h_n * C_t[n] + D[d]*u, with lane-coalesced B_t/C_t loads.
// ---------------------------------------------------------------------------
__global__ void __launch_bounds__(32) k_scan(const float* __restrict__ xconv,
                                             const float* __restrict__ delta,
                                             const float* __restrict__ xdbl,
                                             const float* __restrict__ Alog,
                                             const float* __restrict__ Dp,
                                             float* __restrict__ ycat) {
  const int d = blockIdx.x;               // 0..191
  const int b = blockIdx.y;               // 0..7
  const int lane = threadIdx.x;           // wave32
  float Areg[4], h[4];
#pragma unroll
  for (int j = 0; j < 4; ++j) {
    Areg[j] = -__expf(Alog[(size_t)d * DSTATE + lane + 32 * j]);
    h[j] = 0.f;
  }
  const float Dv = Dp[d];
  const float* urow = xconv + ((size_t)b * DHALF + d) * LSEQ;
  const float* drow = delta + ((size_t)b * DHALF + d) * LSEQ;
  const float* xrow = xdbl + (size_t)b * LSEQ * RPAD;
  float* yrow = ycat + ((size_t)b * DINNER + d) * LSEQ;

  for (int t = 0; t < LSEQ; ++t) {
    const float u  = urow[t];
    const float dt = drow[t];
    const float* xr = xrow + (size_t)t * RPAD;
    __builtin_prefetch(xr + 8 * RPAD, 0, 1);     // global_prefetch_b8
    const float dtu = dt * u;
    float acc = 0.f;
#pragma unroll
    for (int j = 0; j < 4; ++j) {
      const float Bt = xr[DTRANK + lane + 32 * j];
      const float Ct = xr[DTRANK + DSTATE + lane + 32 * j];
      h[j] = fmaf(__expf(dt * Areg[j]), h[j], dtu * Bt);
      acc = fmaf(h[j], Ct, acc);
    }
#pragma unroll
    for (int off = 16; off > 0; off >>= 1)
      acc += __shfl_xor(acc, off, 32);
    if (lane == 0) yrow[t] = fmaf(Dv, u, acc);
  }
}

// ---------------------------------------------------------------------------
// K6: out = ycat^T @ W_out^T, serpentine scatter on store. K = 384.
// ---------------------------------------------------------------------------
__global__ void __launch_bounds__(128) k_gemm_out(const float* __restrict__ ycat,
                                                  const float* __restrict__ Wout,
                                                  float* __restrict__ out) {
  const int lane = threadIdx.x & 31;
  const int wv   = threadIdx.x >> 5;
  const int tm   = blockIdx.x * 4 + wv;   // 0..11 (M = 192)
  const int tn   = blockIdx.y;            // 0..63
  const int b    = blockIdx.z;
  const int m = lane & 15, kh = lane >> 4;
  const int arow = tm * 16 + m;
  const int l = tn * 16 + m;
  const float* bbase = ycat + ((size_t)b * DINNER) * LSEQ + l;

  v8f acc = {0.f, 0.f, 0.f, 0.f, 0.f, 0.f, 0.f, 0.f};
  for (int k = 0; k < DINNER; k += 4) {
    const int kk = k + 2 * kh;
    v2f a = *(const v2f*)(Wout + (size_t)arow * DINNER + kk);
    v2f bf;
    bf.x = bbase[(size_t)kk * LSEQ];
    bf.y = bbase[(size_t)(kk + 1) * LSEQ];
    acc = wmma_f32_step(a, bf, acc);
  }
  const int hr = l >> 5, wc = l & 31;
  const int wdst = (hr & 1) ? (31 - wc) : wc;   // final serpentine restore
  float* dst = out + ((size_t)b * DMODEL) * LSEQ + hr * WCOLS + wdst;
  const int mofs = 8 * kh;
#pragma unroll
  for (int v = 0; v < 8; ++v)
    dst[(size_t)(tm * 16 + v + mofs) * LSEQ] = acc[v];
}

// ---------------------------------------------------------------------------
extern "C" void kernel_launch(void* const* d_in, const int* in_sizes, int n_in,
                              void* d_out, int out_size, void* d_ws, size_t ws_size,
                              hipStream_t stream) {
  (void)in_sizes; (void)n_in; (void)out_size; (void)ws_size;
  const float* hs   = (const float*)d_in[0];
  const float* Win  = (const float*)d_in[1];
  const float* cwx  = (const float*)d_in[2];
  const float* cwz  = (const float*)d_in[3];
  const float* Wxp  = (const float*)d_in[4];
  const float* Wdt  = (const float*)d_in[5];
  const float* bdt  = (const float*)d_in[6];
  const float* Alog = (const float*)d_in[7];
  const float* Dp   = (const float*)d_in[8];
  const float* Wout = (const float*)d_in[9];
  float* out = (float*)d_out;

  float* ws = (float*)d_ws;               // needs 9,437,184 floats = 37.75 MB
  float* xz    = ws;                      // (B, 384, L)   3,145,728
  float* xconv = ws + 3145728;            // (B, 192, L)   1,572,864
  float* ycat  = ws + 4718592;            // (B, 384, L)   3,145,728
  float* delta = ws + 7864320;            // (B, 192, L)   1,572,864
  float* xdbl  = xz;                      // reuse: xz dead after K2; (B,L,272)

  k_gemm_in  <<<dim3(6, 64, BATCH), 128, 0, stream>>>(hs, Win, xz);
  k_conv_silu<<<dim3((BATCH * DINNER * LSEQ) / 256), 256, 0, stream>>>(xz, cwx, cwz, xconv, ycat);
  k_gemm_xproj<<<dim3(17, 64, BATCH), 32, 0, stream>>>(xconv, Wxp, xdbl);
  k_dt_delta <<<dim3((BATCH * DHALF * LSEQ) / 256), 256, 0, stream>>>(xdbl, Wdt, bdt, delta);
  k_scan     <<<dim3(DHALF, BATCH), 32, 0, stream>>>(xconv, delta, xdbl, Alog, Dp, ycat);
  k_gemm_out <<<dim3(3, 64, BATCH), 128, 0, stream>>>(ycat, Wout, out);
}